// ROIPooling_85134841741854
// MI455X (gfx1250) — compile-verified
//
#include <hip/hip_runtime.h>
#include <math.h>
#include <stdint.h>

// ---------------------------------------------------------------------------
// ROI max-pooling for MI455X (gfx1250).
//   feature_map: [B=2, H=64, W=64, C=128] f32
//   rois:        [B=2, N=96, 4] f32  (x_center, y_center, width, height)
//   out:         [B, N, PH=7, PW=7, C=128] f32
//
// One block per (roi, bin-row). Block = (32, 7) = 7 wave32s:
//   lane (threadIdx.x, 0..31) -> channel group of 4 floats (float4), 32*4 = C
//   warp (threadIdx.y, 0..6)  -> pooled bin column j
// Rows of the ROI region are staged into LDS with the gfx1250 async
// global->LDS data mover, double-buffered: row k+1's DMA overlaps row k's
// reduction, fenced with constant-bound s_wait_asynccnt (async loads
// complete in order) + workgroup barriers.
// ---------------------------------------------------------------------------

#define PHB 7
#define PWB 7
#define BB  2
#define NN  96
#define HH  64
#define WW  64
#define CC  128
#define C4  (CC / 4)
#define MAXWPIX 40                 // sizes <= 40 -> region length = 2*floor(sz/2) <= 40
#define NTHREADS (32 * PWB)        // 224 = 7 wave32s
#define ISSUES 6                   // 6 * 224 * 16B = 21504 B >= 40*128*4 = 20480 B

#if __has_builtin(__builtin_amdgcn_global_load_async_to_lds_b128)
#define HAVE_ASYNC_LDS 1
#else
#define HAVE_ASYNC_LDS 0
#endif

typedef int v4i __attribute__((ext_vector_type(4)));
typedef __attribute__((address_space(1))) v4i glb_v4i;   // global int4
typedef __attribute__((address_space(3))) v4i lds_v4i;   // LDS int4

__device__ __forceinline__ int imin(int a, int b) { return a < b ? a : b; }
__device__ __forceinline__ int imax(int a, int b) { return a > b ? a : b; }

// Replicates the reference _bin_index per-axis semantics:
//   half  = floor(size/2)            (float floor)
//   start = int(center - half)       (trunc toward zero, matches astype(int32))
//   end   = int(center + half)
//   len   = end - start
//   step  = max(len // nbins, 1)
__device__ __forceinline__ void bin_axis(float center, float size, int nbins,
                                         int& start, int& len, int& step) {
  float half = floorf(size * 0.5f);
  start = (int)(center - half);
  int end = (int)(center + half);
  len = end - start;
  step = (len > 0) ? (len / nbins) : 0;
  if (step < 1) step = 1;
}

#if HAVE_ASYNC_LDS
// Issue a fixed number (ISSUES) of async b128 global->LDS copies per thread.
// Tail issues redundantly re-copy the last chunk (same bytes, same dest) so
// every wave's ASYNCcnt contribution is exactly ISSUES -> constant waits.
__device__ __forceinline__ void stage_row_async(const float* __restrict__ grow,
                                                float* __restrict__ dst,
                                                int chunks, int tid) {
#pragma unroll
  for (int it = 0; it < ISSUES; ++it) {
    int p = tid + it * NTHREADS;
    int pc = (p < chunks) ? p : (chunks - 1);
    __builtin_amdgcn_global_load_async_to_lds_b128(
        (glb_v4i*)(uintptr_t)(grow + 4 * pc),
        (lds_v4i*)(uint32_t)(uintptr_t)(dst + 4 * pc),
        0, 0);
  }
}
__device__ __forceinline__ void wait_async_le6() {
#if __has_builtin(__builtin_amdgcn_s_wait_asynccnt)
  __builtin_amdgcn_s_wait_asynccnt(ISSUES);
#else
  asm volatile("s_wait_asynccnt 6" ::: "memory");
#endif
}
__device__ __forceinline__ void wait_async_le0() {
#if __has_builtin(__builtin_amdgcn_s_wait_asynccnt)
  __builtin_amdgcn_s_wait_asynccnt(0);
#else
  asm volatile("s_wait_asynccnt 0" ::: "memory");
#endif
}
#endif

__global__ __launch_bounds__(NTHREADS)
void roi_maxpool_gfx1250(const float* __restrict__ fm,
                         const float* __restrict__ rois,
                         float* __restrict__ out) {
  __shared__ __align__(16) float sbuf[2][MAXWPIX * CC];  // 2 x 20 KB row stages

  const int roi  = blockIdx.x;        // b*N + n
  const int bi   = blockIdx.y;        // pooled bin row i
  const int b    = roi / NN;
  const int lane = threadIdx.x;       // channel group
  const int j    = threadIdx.y;       // pooled bin col (one wave32 each)
  const int tid  = j * 32 + lane;

  // ROI params (uniform across block)
  const float4 r4 = ((const float4*)rois)[roi];  // x=xc, y=yc, z=w, w=h
  int xs, xlen, xstep; bin_axis(r4.x, r4.z, PWB, xs, xlen, xstep);
  int ys, ylen, ystep; bin_axis(r4.y, r4.w, PHB, ys, ylen, ystep);

  // Rows belonging to bin row `bi` (block-uniform -> barriers never diverge).
  int rlo = bi * ystep;
  int rhi = (bi == PHB - 1) ? ylen : imin((bi + 1) * ystep, ylen);
  int ylo = imax(ys + rlo, 0);
  int yhi = imin(ys + rhi, HH);
  const int nrows = yhi - ylo;

  // Staged column segment = whole ROI width, clamped to the image.
  int xclo = imax(xs, 0);
  int xchi = imin(xs + xlen, WW);
  int wpix = xchi - xclo;
  if (wpix > MAXWPIX) wpix = MAXWPIX;  // defensive (inputs guarantee <= 40)

  // My bin column's region-relative column range, clamped to the staged span.
  int clo = j * xstep;
  int chi = (j == PWB - 1) ? xlen : imin((j + 1) * xstep, xlen);
  int cl  = imax(clo, xclo - xs);
  int ch  = imin(chi, xchi - xs);

  float4 acc = make_float4(-INFINITY, -INFINITY, -INFINITY, -INFINITY);

  const int chunks = wpix * C4;  // float4 chunks per staged row (<= 1280)

  if (wpix > 0 && nrows > 0) {
    const size_t rowstride = (size_t)WW * CC;
    const float* g0 = fm + (((size_t)(b * HH + ylo)) * WW + xclo) * CC;

#if HAVE_ASYNC_LDS
    // Prime the 2-deep pipeline.
    stage_row_async(g0, sbuf[0], chunks, tid);
    if (nrows > 1) stage_row_async(g0 + rowstride, sbuf[1], chunks, tid);

    for (int k = 0; k < nrows; ++k) {
      // Async loads complete in order: <=ISSUES outstanding => row k landed,
      // while row k+1's DMA may still be in flight.
      if (k + 1 < nrows) wait_async_le6();
      else               wait_async_le0();
      __syncthreads();                       // all waves' row-k data visible

      const float4* sp = (const float4*)sbuf[k & 1];
      for (int c = cl; c < ch; ++c) {
        int s = (xs + c) - xclo;             // staged pixel index
        float4 v = sp[s * C4 + lane];
        acc.x = fmaxf(acc.x, v.x);
        acc.y = fmaxf(acc.y, v.y);
        acc.z = fmaxf(acc.z, v.z);
        acc.w = fmaxf(acc.w, v.w);
      }
      __syncthreads();                       // everyone done reading buf k&1

      if (k + 2 < nrows)                     // refill the buffer just drained
        stage_row_async(g0 + (size_t)(k + 2) * rowstride, sbuf[k & 1], chunks, tid);
    }
#else
    // Fallback staging: coalesced b128 loads + LDS stores, single buffer.
    for (int k = 0; k < nrows; ++k) {
      const float* grow = g0 + (size_t)k * rowstride;
      for (int p = tid; p < chunks; p += NTHREADS)
        ((float4*)sbuf[0])[p] = ((const float4*)grow)[p];
      __syncthreads();
      const float4* sp = (const float4*)sbuf[0];
      for (int c = cl; c < ch; ++c) {
        int s = (xs + c) - xclo;
        float4 v = sp[s * C4 + lane];
        acc.x = fmaxf(acc.x, v.x);
        acc.y = fmaxf(acc.y, v.y);
        acc.z = fmaxf(acc.z, v.z);
        acc.w = fmaxf(acc.w, v.w);
      }
      __syncthreads();
    }
#endif
  }

  // out[((roi*PH + i)*PW + j)*C + 4*lane .. +3]; empty bins stay -inf
  // (matches segment_max identity in the reference).
  size_t obin = ((size_t)roi * PHB + bi) * PWB + j;
  ((float4*)out)[obin * C4 + lane] = acc;
}

extern "C" void kernel_launch(void* const* d_in, const int* in_sizes, int n_in,
                              void* d_out, int out_size, void* d_ws, size_t ws_size,
                              hipStream_t stream) {
  const float* fm   = (const float*)d_in[0];   // feature_map [2,64,64,128]
  const float* rois = (const float*)d_in[1];   // rois        [2,96,4]
  float* out        = (float*)d_out;           // [2,96,7,7,128]
  (void)in_sizes; (void)n_in; (void)out_size; (void)d_ws; (void)ws_size;

  dim3 grid(BB * NN, PHB);      // (192, 7): one block per (roi, bin-row)
  dim3 block(32, PWB);          // 7 wave32s
  hipLaunchKernelGGL(roi_maxpool_gfx1250, grid, block, 0, stream, fm, rois, out);
}